// ModelNew_32607391711562
// MI455X (gfx1250) — compile-verified
//
#include <hip/hip_runtime.h>
#include <hip/hip_bf16.h>

// 3D Winograd F(2x2x2, 3x3x3) convolution for N=4, Ci=Co=128, D=H=W=32.
//   U2[p][ci/2][co][2]  interleaved ci-pairs -> one b64 A-load per WMMA  (4 MB ws)
//   V[p][ci][t]         t = n*4096+tz*256+ty*16+tx                      (512 MB ws)
//   y = (At (x) At (x) At) . (U[p] @ V[p])   fused epilogue, f32 WMMA,
//   V panels double-buffered in LDS via GLOBAL_LOAD_ASYNC_TO_LDS (ASYNCcnt).

typedef float v2f __attribute__((ext_vector_type(2)));
typedef float v8f __attribute__((ext_vector_type(8)));

typedef __attribute__((address_space(1))) int GInt;   // global (device) int
typedef __attribute__((address_space(3))) int LInt;   // LDS int

#define CI   128
#define CO   128
#define NT   16384           // 4 * 16*16*16 tiles
#define DHW  32
#define VROW 48              // LDS row stride (floats): k-row pairs land on
                             // disjoint bank halves for both 16-lane groups

#if defined(__HIP_DEVICE_COMPILE__) && __has_builtin(__builtin_amdgcn_global_load_async_to_lds_b32)
#define HAS_ASYNC_LDS 1
#else
#define HAS_ASYNC_LDS 0
#endif

// ---------------------------------------------------------------------------
// Kernel 1: weight transform into paired layout
//   U2[((p*64 + ci/2)*128 + co)*2 + (ci&1)] = sum G[a,d]G[b,e]G[c,f] w[co][ci][def]
// G = [[1,0,0],[.5,.5,.5],[.5,-.5,.5],[0,0,1]]
// ---------------------------------------------------------------------------
__global__ __launch_bounds__(256)
void wg_weight_transform(const float* __restrict__ w, float* __restrict__ U2) {
    const int gid = blockIdx.x * 256 + threadIdx.x;     // 128*128 threads
    const int co = gid & 127;
    const int ci = gid >> 7;
    const float* wp = w + ((size_t)co * CI + ci) * 27;

    float s[4][4][4];
    #pragma unroll
    for (int e = 0; e < 3; ++e)
        #pragma unroll
        for (int f = 0; f < 3; ++f) {
            float g0 = wp[(0 * 3 + e) * 3 + f];
            float g1 = wp[(1 * 3 + e) * 3 + f];
            float g2 = wp[(2 * 3 + e) * 3 + f];
            s[0][e][f] = g0;
            s[1][e][f] = 0.5f * (g0 + g1 + g2);
            s[2][e][f] = 0.5f * (g0 - g1 + g2);
            s[3][e][f] = g2;
        }
    #pragma unroll
    for (int a = 0; a < 4; ++a)
        #pragma unroll
        for (int f = 0; f < 3; ++f) {
            float g0 = s[a][0][f], g1 = s[a][1][f], g2 = s[a][2][f];
            s[a][0][f] = g0;
            s[a][1][f] = 0.5f * (g0 + g1 + g2);
            s[a][2][f] = 0.5f * (g0 - g1 + g2);
            s[a][3][f] = g2;
        }
    #pragma unroll
    for (int a = 0; a < 4; ++a)
        #pragma unroll
        for (int b = 0; b < 4; ++b) {
            float g0 = s[a][b][0], g1 = s[a][b][1], g2 = s[a][b][2];
            s[a][b][0] = g0;
            s[a][b][1] = 0.5f * (g0 + g1 + g2);
            s[a][b][2] = 0.5f * (g0 - g1 + g2);
            s[a][b][3] = g2;
        }
    const size_t pairOff = ((size_t)(ci >> 1) * 128 + co) * 2 + (ci & 1);
    #pragma unroll
    for (int a = 0; a < 4; ++a)
        #pragma unroll
        for (int b = 0; b < 4; ++b)
            #pragma unroll
            for (int c = 0; c < 4; ++c) {
                int p = (a * 4 + b) * 4 + c;
                U2[(size_t)p * (64 * 128 * 2) + pairOff] = s[a][b][c];
            }
}

// ---------------------------------------------------------------------------
// Kernel 2: input transform  V[p][ci][t] = (Bt (x) Bt (x) Bt) . x-patch
// Bt = [[1,0,-1,0],[0,1,1,0],[0,-1,1,0],[0,1,0,-1]], patch origin 2*tile-1 (pad 1)
// ---------------------------------------------------------------------------
__global__ __launch_bounds__(256)
void wg_input_transform(const float* __restrict__ x, float* __restrict__ V) {
    const int gid = blockIdx.x * 256 + threadIdx.x;     // 128 * 16384 threads
    const int ci = gid >> 14;
    const int t  = gid & (NT - 1);
    const int n  = t >> 12;
    const int rz = t & 4095;
    const int tz = rz >> 8;
    const int ty = (rz >> 4) & 15;
    const int tx = rz & 15;
    const int d0 = 2 * tz - 1, h0 = 2 * ty - 1, w0 = 2 * tx - 1;
    const float* xp = x + (((size_t)n * CI + ci) << 15);   // * 32^3

    float s[4][4][4];
    #pragma unroll
    for (int e = 0; e < 4; ++e) {
        const int yy = h0 + e;
        const bool vy = (unsigned)yy < DHW;
        #pragma unroll
        for (int f = 0; f < 4; ++f) {
            const int xx = w0 + f;
            const bool vx = (unsigned)xx < DHW;
            float m0 = 0.f, m1 = 0.f, m2 = 0.f, m3 = 0.f;
            if (vy && vx) {
                const int base = yy * DHW + xx;
                int z;
                z = d0 + 0; m0 = ((unsigned)z < DHW) ? xp[z * (DHW * DHW) + base] : 0.f;
                z = d0 + 1; m1 = ((unsigned)z < DHW) ? xp[z * (DHW * DHW) + base] : 0.f;
                z = d0 + 2; m2 = ((unsigned)z < DHW) ? xp[z * (DHW * DHW) + base] : 0.f;
                z = d0 + 3; m3 = ((unsigned)z < DHW) ? xp[z * (DHW * DHW) + base] : 0.f;
            }
            s[0][e][f] = m0 - m2;
            s[1][e][f] = m1 + m2;
            s[2][e][f] = m2 - m1;
            s[3][e][f] = m1 - m3;
        }
    }
    #pragma unroll
    for (int a = 0; a < 4; ++a)
        #pragma unroll
        for (int f = 0; f < 4; ++f) {
            float m0 = s[a][0][f], m1 = s[a][1][f], m2 = s[a][2][f], m3 = s[a][3][f];
            s[a][0][f] = m0 - m2;
            s[a][1][f] = m1 + m2;
            s[a][2][f] = m2 - m1;
            s[a][3][f] = m1 - m3;
        }
    #pragma unroll
    for (int a = 0; a < 4; ++a)
        #pragma unroll
        for (int b = 0; b < 4; ++b) {
            float m0 = s[a][b][0], m1 = s[a][b][1], m2 = s[a][b][2], m3 = s[a][b][3];
            s[a][b][0] = m0 - m2;
            s[a][b][1] = m1 + m2;
            s[a][b][2] = m2 - m1;
            s[a][b][3] = m1 - m3;
        }
    float* Vo = V + (size_t)ci * NT + t;
    #pragma unroll
    for (int a = 0; a < 4; ++a)
        #pragma unroll
        for (int b = 0; b < 4; ++b)
            #pragma unroll
            for (int c = 0; c < 4; ++c)
                Vo[(size_t)((a * 4 + b) * 4 + c) * (CI * NT)] = s[a][b][c];
}

// ---------------------------------------------------------------------------
// Kernel 3 helpers: async (or sync fallback) staging of one V panel into LDS
// ---------------------------------------------------------------------------
__device__ __forceinline__ void stage_panel(const float* __restrict__ Vp,
                                            float* dst, int tid) {
    #pragma unroll
    for (int i = 0; i < 8; ++i) {
        const int e  = i * 512 + tid;       // 4096 elements = 128ci x 32 tiles
        const int ci = e >> 5;
        const int tl = e & 31;
#if HAS_ASYNC_LDS
        __builtin_amdgcn_global_load_async_to_lds_b32(
            (GInt*)(Vp + (size_t)ci * NT + tl),
            (LInt*)(dst + ci * VROW + tl),
            0, 0);
#else
        dst[ci * VROW + tl] = Vp[(size_t)ci * NT + tl];
#endif
    }
}

__device__ __forceinline__ void wait_async_lds() {
#if HAS_ASYNC_LDS
 #if __has_builtin(__builtin_amdgcn_s_wait_asynccnt)
    __builtin_amdgcn_s_wait_asynccnt(0);
 #else
    asm volatile("s_wait_asynccnt 0x0" ::: "memory");
 #endif
#endif
}

// ---------------------------------------------------------------------------
// Kernel 3: per-point GEMM (f32 WMMA 16x16x4) + fused inverse transform.
// 512 threads = 16 waves; wave (cg,tg): co rows [cg*16,+16), tiles [tg*16,+16).
// V panels double-buffered in LDS; A loaded as float2 k-row pairs (b64).
// ---------------------------------------------------------------------------
__global__ __launch_bounds__(512)
void wg_gemm_fused(const float* __restrict__ U, const float* __restrict__ V,
                   float* __restrict__ y) {
    __shared__ float Vl[2][CI * VROW];          // 2 x 24 KB

    const int tid   = threadIdx.x;
    const int lane  = tid & 31;
    const int wave  = tid >> 5;
    const int l16   = lane & 15;
    const int hi    = lane >> 4;              // 0: K rows 0/1, 1: K rows 2/3
    const int coBase = (wave & 7) * 16;
    const int tg     = wave >> 3;             // tile half of the 32-tile block
    const int blockTile = blockIdx.x * 32;

    const int t  = blockTile + tg * 16 + l16;
    const int n  = t >> 12;
    const int rz = t & 4095;
    const int tz = rz >> 8;
    const int ty = (rz >> 4) & 15;
    const int tx = rz & 15;

    // At = [[1,1,1,0],[0,1,-1,-1]]
    const float At0[4] = {1.f, 1.f, 1.f, 0.f};
    const float At1[4] = {0.f, 1.f, -1.f, -1.f};

    const v2f* U2 = (const v2f*)U;            // [p][ci/2][co] float2

    float Y[2][2][2][8];
    #pragma unroll
    for (int k = 0; k < 2; ++k)
        #pragma unroll
        for (int l = 0; l < 2; ++l)
            #pragma unroll
            for (int m = 0; m < 2; ++m)
                #pragma unroll
                for (int r = 0; r < 8; ++r) Y[k][l][m][r] = 0.f;

    // prologue: stage panel 0 into buffer 0
    stage_panel(V + blockTile, &Vl[0][0], tid);

    for (int a = 0; a < 4; ++a) {
        float Q[2][2][8];
        #pragma unroll
        for (int l = 0; l < 2; ++l)
            #pragma unroll
            for (int m = 0; m < 2; ++m)
                #pragma unroll
                for (int r = 0; r < 8; ++r) Q[l][m][r] = 0.f;

        for (int b = 0; b < 4; ++b) {
            float P[2][8];
            #pragma unroll
            for (int m = 0; m < 2; ++m)
                #pragma unroll
                for (int r = 0; r < 8; ++r) P[m][r] = 0.f;

            for (int c = 0; c < 4; ++c) {
                const int p   = (a * 4 + b) * 4 + c;
                const int buf = p & 1;

                // panel p is resident after wait + barrier
                wait_async_lds();
                __syncthreads();
                if (p < 63)                    // overlap: stage next panel
                    stage_panel(V + (size_t)(p + 1) * (CI * NT) + blockTile,
                                &Vl[buf ^ 1][0], tid);

                const v2f* Up = U2 + (size_t)p * (64 * 128) + coBase + l16;
                __builtin_prefetch(U2 + (size_t)((p + 1) & 63) * (64 * 128) +
                                   coBase + l16, 0, 0);
                const float* Vb = &Vl[buf][0] + tg * 16 + l16;

                // K=128 GEMM: 32 chained v_wmma_f32_16x16x4_f32
                v8f acc = {};
                #pragma unroll 4
                for (int k = 0; k < CI; k += 4) {
                    const int k0 = k + 2 * hi;          // ISA f32 A/B K layout
                    v2f av = Up[(size_t)((k >> 1) + hi) * 128];
                    v2f bv;
                    bv.x = Vb[(k0 + 0) * VROW];
                    bv.y = Vb[(k0 + 1) * VROW];
                    acc = __builtin_amdgcn_wmma_f32_16x16x4_f32(
                        false, av, false, bv, (short)0, acc, false, false);
                }

                // inverse transform, axis c:  P[m] += At[m][c] * M
                const float a0 = At0[c], a1 = At1[c];
                #pragma unroll
                for (int r = 0; r < 8; ++r) {
                    P[0][r] += a0 * acc[r];
                    P[1][r] += a1 * acc[r];
                }
            }
            const float b0 = At0[b], b1 = At1[b];
            #pragma unroll
            for (int m = 0; m < 2; ++m)
                #pragma unroll
                for (int r = 0; r < 8; ++r) {
                    Q[0][m][r] += b0 * P[m][r];
                    Q[1][m][r] += b1 * P[m][r];
                }
        }
        const float c0 = At0[a], c1 = At1[a];
        #pragma unroll
        for (int l = 0; l < 2; ++l)
            #pragma unroll
            for (int m = 0; m < 2; ++m)
                #pragma unroll
                for (int r = 0; r < 8; ++r) {
                    Y[0][l][m][r] += c0 * Q[l][m][r];
                    Y[1][l][m][r] += c1 * Q[l][m][r];
                }
    }

    // store 2x2x2 outputs for 8 co rows (C-layout: reg r -> co r / r+8)
    #pragma unroll
    for (int k = 0; k < 2; ++k)
        #pragma unroll
        for (int l = 0; l < 2; ++l)
            #pragma unroll
            for (int m = 0; m < 2; ++m)
                #pragma unroll
                for (int r = 0; r < 8; ++r) {
                    const int co = coBase + r + hi * 8;
                    const size_t idx =
                        ((((size_t)n * CO + co) * DHW + (2 * tz + k)) * DHW +
                         (2 * ty + l)) * DHW + (2 * tx + m);
                    y[idx] = Y[k][l][m][r];
                }
}

// ---------------------------------------------------------------------------
// Needs ws >= 4 MB (U2) + 512 MB (V) = 516 MB.
// ---------------------------------------------------------------------------
extern "C" void kernel_launch(void* const* d_in, const int* in_sizes, int n_in,
                              void* d_out, int out_size, void* d_ws, size_t ws_size,
                              hipStream_t stream) {
    const float* x = (const float*)d_in[0];     // (4,128,32,32,32)
    const float* w = (const float*)d_in[1];     // (128,128,3,3,3)
    float* y = (float*)d_out;                   // (4,128,32,32,32)
    float* U = (float*)d_ws;                    // 64 * 64 * 128 float2
    float* V = U + (size_t)64 * CI * CO;        // 64 * 128 * 16384

    wg_weight_transform<<<(CI * CO) / 256, 256, 0, stream>>>(w, U);
    wg_input_transform<<<(CI * NT) / 256, 256, 0, stream>>>(x, V);
    wg_gemm_fused<<<NT / 32, 512, 0, stream>>>(U, V, y);
}